// PConv_9165460209718
// MI455X (gfx1250) — compile-verified
//
#include <hip/hip_runtime.h>

typedef __attribute__((ext_vector_type(2))) float v2f;
typedef __attribute__((ext_vector_type(8))) float v8f;

// Problem constants (match reference)
constexpr int BB    = 2;
constexpr int NN    = 50000;
constexpr int MM    = 50000;
constexpr int KK    = 16;      // neighbors = WMMA K dim (4 steps of K=4)
constexpr int C_IN  = 64;
constexpr int C_ADD = 3;
constexpr int C_MID = 16;      // WMMA N dim
constexpr int C_TOT = C_IN + C_ADD;   // 67 channels -> 5 tiles of 16 (padded)
constexpr int C_OUT = C_TOT * C_MID;  // 1072 floats per point
constexpr int NPTS  = BB * MM;
constexpr int WAVES_PER_BLOCK = 8;    // 256 threads = 8 wave32

__global__ __launch_bounds__(256) void pconv_wmma_f32(
    const float* __restrict__ feat,   // [B, N, 64]
    const int*   __restrict__ nbr,    // [B, M, 16]
    const float* __restrict__ wn,     // [B, M, 16, 16]
    const float* __restrict__ addf,   // [B, M, 16, 3]
    float*       __restrict__ out)    // [B, M, 1072]
{
    const int wave = threadIdx.x >> 5;
    const int lane = threadIdx.x & 31;
    const int p    = blockIdx.x * WAVES_PER_BLOCK + wave;   // one point per wave
    if (p >= NPTS) return;   // whole-wave guard: EXEC all-1s inside

    const int b    = p / MM;
    const int half = lane >> 4;   // 0: K pair {0,1}, 1: K pair {2,3} within a 4-slice
    const int ln   = lane & 15;   // A: channel-in-tile; B/D: output column w

    const float* featB = feat + (size_t)b * NN * C_IN;
    const int*   idx   = nbr  + (size_t)p * KK;
    const float* wnP   = wn   + (size_t)p * KK * C_MID;
    const float* addP  = addf + (size_t)p * KK * C_ADD;
    float*       outP  = out  + (size_t)p * C_OUT;

    v8f acc[5] = {};   // 5 channel tiles of 16x16 f32 accumulators

    #pragma unroll
    for (int s = 0; s < 4; ++s) {              // K = 16 neighbors, 4 per WMMA
        const int k0 = 4 * s + 2 * half;       // this lane's K pair in the slice

        // B-matrix slice wn[k0..k0+1, ln] — hoisted, reused by all 5 tiles.
        v2f bm;
        bm.x = wnP[(size_t)k0 * C_MID + ln];
        bm.y = wnP[(size_t)(k0 + 1) * C_MID + ln];

        // Gathered neighbor feature rows for this lane's two K values.
        const float* r0 = featB + (size_t)idx[k0]     * C_IN;
        const float* r1 = featB + (size_t)idx[k0 + 1] * C_IN;

        #pragma unroll
        for (int t = 0; t < 5; ++t) {          // 67 channels -> 5 row tiles
            v2f am;
            if (t < 4) {
                const int c = 16 * t + ln;     // gathered input features
                am.x = r0[c];
                am.y = r1[c];
            } else {                           // additional features, zero-padded
                am.x = (ln < C_ADD) ? addP[(size_t)k0 * C_ADD + ln]       : 0.0f;
                am.y = (ln < C_ADD) ? addP[(size_t)(k0 + 1) * C_ADD + ln] : 0.0f;
            }
            // D = A(16x4) x B(4x16) + C, full fp32
            acc[t] = __builtin_amdgcn_wmma_f32_16x16x4_f32(
                false, am, false, bm, (short)0, acc[t], false, false);
        }
    }

    // D layout: VGPR r, lanes 0-15 -> M=r, lanes 16-31 -> M=8+r; column = ln.
    #pragma unroll
    for (int t = 0; t < 5; ++t) {
        #pragma unroll
        for (int r = 0; r < 8; ++r) {
            const int c = 16 * t + 8 * half + r;
            if (c < C_TOT) outP[(size_t)c * C_MID + ln] = acc[t][r];
        }
    }
}

extern "C" void kernel_launch(void* const* d_in, const int* in_sizes, int n_in,
                              void* d_out, int out_size, void* d_ws, size_t ws_size,
                              hipStream_t stream) {
    const float* feat = (const float*)d_in[0];   // input_features  [B,N,64] f32
    const int*   nbr  = (const int*)  d_in[1];   // neighbor_inds   [B,M,16] int
    const float* wn   = (const float*)d_in[2];   // weightnet       [B,M,16,16] f32
    const float* addf = (const float*)d_in[3];   // additional      [B,M,16,3] f32
    float*       out  = (float*)d_out;           // [B,M,1072] f32

    const int blocks = (NPTS + WAVES_PER_BLOCK - 1) / WAVES_PER_BLOCK;
    pconv_wmma_f32<<<blocks, 256, 0, stream>>>(feat, nbr, wn, addf, out);
}